// CE_41884521071185
// MI455X (gfx1250) — compile-verified
//
#include <hip/hip_runtime.h>
#include <hip/hip_bf16.h>
#include <math.h>

// Problem constants (B,S,C,K) = (16,256,32768,8)
#define BB 16
#define SS 256
#define CC 32768
#define KK 8
#define NROWS (BB * SS)          // 4096 rows
#define TPB 256                  // 8 wave32s per block
#define EPT (CC / TPB)           // 128 elements per thread
#define NWAVES (TPB / 32)        // 8

// ---------------------------------------------------------------------------
// Kernel 0: zero the histogram region of the workspace (re-init every call)
// ---------------------------------------------------------------------------
__global__ __launch_bounds__(256) void ce_init_hist(float* hist) {
  int i = blockIdx.x * blockDim.x + threadIdx.x;
  if (i < CC) hist[i] = 0.0f;
}

// block tree-reduce helper (max or sum) through LDS scratch
__device__ __forceinline__ float blk_reduce(float v, float* red, int tid, bool do_max) {
  red[tid] = v;
  __syncthreads();
  for (int s = TPB / 2; s > 0; s >>= 1) {
    if (tid < s) red[tid] = do_max ? fmaxf(red[tid], red[tid + s]) : (red[tid] + red[tid + s]);
    __syncthreads();
  }
  float r = red[0];
  __syncthreads();
  return r;
}

// ---------------------------------------------------------------------------
// Kernel 1: one block per row (32768 f32 = 128 KB; whole row fits in LDS).
//   - CDNA5 async global->LDS DMA: HBM read exactly once
//   - logsumexp (max pass + exp-sum pass) out of LDS with b128 reads
//   - top-8: per-thread sorted list -> per-wave shuffle merge -> 1-wave final
//   - per-row loss / match to workspace slots (deterministic reduce later)
// ---------------------------------------------------------------------------
__global__ __launch_bounds__(TPB) void ce_row_kernel(
    const float* __restrict__ logits, const int* __restrict__ tgt,
    float* __restrict__ out, float* __restrict__ row_loss,
    float* __restrict__ row_match, float* __restrict__ hist) {
  __shared__ __align__(16) float row[CC];      // 131072 B
  __shared__ float cval[NWAVES * KK];          //    256 B (64 candidates)
  __shared__ int   cidx[NWAVES * KK];          //    256 B
  __shared__ float red[TPB];                   //   1024 B  => ~130 KB total

  const int tid  = threadIdx.x;
  const int lane = tid & 31;
  const int wv   = tid >> 5;
  const int r    = blockIdx.x;

  // ---- Stage the row into LDS with CDNA5 async global->LDS loads ----------
  // 16 B per lane per op; 256 lanes cover 4 KB/iter; 32 iters = 128 KB.
  {
    const uint64_t gbase = (uint64_t)(uintptr_t)(logits + (size_t)r * CC);
    const uint32_t lbase = (uint32_t)(uintptr_t)&row[0];  // low 32 bits of flat
                                                          // shared ptr = LDS offset
    for (int it = 0; it < (CC * 4) / (TPB * 16); ++it) {
      uint32_t boff = (uint32_t)(it * TPB + tid) * 16u;
      uint32_t l = lbase + boff;
      uint64_t g = gbase + (uint64_t)boff;
      asm volatile("global_load_async_to_lds_b128 %0, %1, off"
                   :: "v"(l), "v"(g) : "memory");
    }
    asm volatile("s_wait_asynccnt 0" ::: "memory");  // drain ASYNCcnt (per wave)
  }
  __syncthreads();

  // ---- Pass A: per-thread sorted top-8 over 128 elements (b128 LDS reads) -
  float tv[KK];
  int   ti[KK];
#pragma unroll
  for (int k = 0; k < KK; ++k) { tv[k] = -INFINITY; ti[k] = 0x7FFFFFFF; }

  const float4* rowv4 = (const float4*)row;
  for (int i = 0; i < EPT / 4; ++i) {
    int j = i * TPB + tid;                  // float4 index, 0..8191
    float4 v4 = rowv4[j];
#pragma unroll
    for (int c = 0; c < 4; ++c) {
      float v = (c == 0) ? v4.x : (c == 1) ? v4.y : (c == 2) ? v4.z : v4.w;
      int col = j * 4 + c;
      if (v > tv[KK - 1]) {                 // strict '>' keeps lower idx on tie
#pragma unroll
        for (int k = KK - 1; k >= 1; --k) {
          bool up  = v > tv[k - 1];
          bool ins = !up && (v > tv[k]);
          float ntv = up ? tv[k - 1] : (ins ? v : tv[k]);
          int   nti = up ? ti[k - 1] : (ins ? col : ti[k]);
          tv[k] = ntv; ti[k] = nti;
        }
        if (v > tv[0]) { tv[0] = v; ti[0] = col; }
      }
    }
  }
  float lmax = tv[0];                       // thread-local max

  // ---- logsumexp: row max, then sum of exp(x - max) -----------------------
  float m = blk_reduce(lmax, red, tid, true);
  float s = 0.0f;
  for (int i = 0; i < EPT / 4; ++i) {
    float4 v4 = rowv4[i * TPB + tid];
    s += __expf(v4.x - m) + __expf(v4.y - m) + __expf(v4.z - m) + __expf(v4.w - m);
  }
  float ssum = blk_reduce(s, red, tid, false);

  // ---- Stage 1: each wave merges its 256 candidates to a wave-top-8 -------
  // (shuffle-argmax over list heads; winner lane pops its sorted list)
#pragma unroll
  for (int k = 0; k < KK; ++k) {
    float bv = tv[0]; int bi = ti[0]; int bl = lane;
#pragma unroll
    for (int o = 16; o > 0; o >>= 1) {
      float ov = __shfl_down(bv, o, 32);
      int   oi = __shfl_down(bi, o, 32);
      int   ol = __shfl_down(bl, o, 32);
      if (ov > bv || (ov == bv && oi < bi)) { bv = ov; bi = oi; bl = ol; }
    }
    bv = __shfl(bv, 0, 32); bi = __shfl(bi, 0, 32); bl = __shfl(bl, 0, 32);
    if (lane == bl) {                       // pop winner's list head
#pragma unroll
      for (int q = 0; q < KK - 1; ++q) { tv[q] = tv[q + 1]; ti[q] = ti[q + 1]; }
      tv[KK - 1] = -INFINITY; ti[KK - 1] = 0x7FFFFFFF;
    }
    if (lane == 0) { cval[wv * KK + k] = bv; cidx[wv * KK + k] = bi; }
  }
  __syncthreads();

  // ---- Stage 2: one wave merges the 64 wave-winners to the global top-8 ---
  float topv[KK];
  int   topi[KK];
  if (tid < 32) {
    float av = cval[tid];      int ai = cidx[tid];
    float bv2 = cval[tid + 32]; int bi2 = cidx[tid + 32];
    if (bv2 > av || (bv2 == av && bi2 < ai)) {   // keep the better in 'a'
      float tf = av; av = bv2; bv2 = tf;
      int   tn = ai; ai = bi2; bi2 = tn;
    }
#pragma unroll
    for (int k = 0; k < KK; ++k) {
      float bv = av; int bi = ai; int bl = tid;
#pragma unroll
      for (int o = 16; o > 0; o >>= 1) {
        float ov = __shfl_down(bv, o, 32);
        int   oi = __shfl_down(bi, o, 32);
        int   ol = __shfl_down(bl, o, 32);
        if (ov > bv || (ov == bv && oi < bi)) { bv = ov; bi = oi; bl = ol; }
      }
      bv = __shfl(bv, 0, 32); bi = __shfl(bi, 0, 32); bl = __shfl(bl, 0, 32);
      if (tid == bl) { av = bv2; ai = bi2; bv2 = -INFINITY; bi2 = 0x7FFFFFFF; }
      topv[k] = bv; topi[k] = bi;           // all lanes track (lane0 uses it)
    }
  }

  // ---- per-row finalize (lane 0): loss, preds, matches, histogram ---------
  if (tid == 0) {
    float lse = m + __logf(ssum);
    int gts[KK];
    float gsum = 0.0f;
#pragma unroll
    for (int j = 0; j < KK; ++j) {
      gts[j] = tgt[r * KK + j];
      gsum += row[gts[j]];                  // gather straight from LDS
    }
    row_loss[r] = lse - gsum * (1.0f / KK);

    int matches = 0;
#pragma unroll
    for (int j = 0; j < KK; ++j) {
      int p = topi[j];
      (void)topv[j];
      out[1 + r * KK + j] = (float)p;       // preds region of d_out
      bool hit = false;
#pragma unroll
      for (int jj = 0; jj < KK; ++jj) hit = hit || (p == gts[jj]);
      matches += hit ? 1 : 0;
      atomicAdd(&hist[p], 1.0f);            // exact small-int float adds:
    }                                       // order-independent => deterministic
    row_match[r] = (float)matches;
  }
}

// ---------------------------------------------------------------------------
// Kernel 2: deterministic final reductions + percentage histogram
//   d_out layout (f32): [0]=loss, [1..32768]=preds, [32769]=acc,
//                       [32770..65537]=p_fracs
// ---------------------------------------------------------------------------
__global__ __launch_bounds__(256) void ce_finalize(
    const float* __restrict__ row_loss, const float* __restrict__ row_match,
    const float* __restrict__ hist, float* __restrict__ out) {
  __shared__ float red[256];
  const int tid = threadIdx.x;

  float ls = 0.0f, mt = 0.0f;
  for (int k = 0; k < NROWS / 256; ++k) {   // fixed-order partial sums
    ls += row_loss[tid + k * 256];
    mt += row_match[tid + k * 256];
  }
  red[tid] = ls; __syncthreads();
  for (int s = 128; s > 0; s >>= 1) { if (tid < s) red[tid] += red[tid + s]; __syncthreads(); }
  float loss_sum = red[0]; __syncthreads();
  red[tid] = mt; __syncthreads();
  for (int s = 128; s > 0; s >>= 1) { if (tid < s) red[tid] += red[tid + s]; __syncthreads(); }
  float match_sum = red[0]; __syncthreads();

  if (tid == 0) {
    out[0] = loss_sum / (float)NROWS;
    out[1 + NROWS * KK] = match_sum * (100.0f / (float)(NROWS * KK));
  }
  // p_counts.sum() == NROWS*KK == 32768 always; int32 cast truncates.
  for (int i = tid; i < CC; i += 256)
    out[2 + NROWS * KK + i] = (float)(int)(hist[i] * (100.0f / 32768.0f));
}

// ---------------------------------------------------------------------------
extern "C" void kernel_launch(void* const* d_in, const int* in_sizes, int n_in,
                              void* d_out, int out_size, void* d_ws, size_t ws_size,
                              hipStream_t stream) {
  const float* logits = (const float*)d_in[0];   // (16,256,32768) f32
  const int*   tgt    = (const int*)d_in[1];     // (16,256,8) i32
  float* out = (float*)d_out;
  float* ws  = (float*)d_ws;
  float* row_loss  = ws;                 // 4096 f32
  float* row_match = ws + NROWS;         // 4096 f32
  float* hist      = ws + 2 * NROWS;     // 32768 f32

  ce_init_hist<<<CC / 256, 256, 0, stream>>>(hist);
  ce_row_kernel<<<NROWS, TPB, 0, stream>>>(logits, tgt, out, row_loss, row_match, hist);
  ce_finalize<<<1, 256, 0, stream>>>(row_loss, row_match, hist, out);
}